// GaussianSmoother_89739046682806
// MI455X (gfx1250) — compile-verified
//
#include <hip/hip_runtime.h>
#include <cstdint>

namespace {
constexpr int kB      = 2;
constexpr int kN      = 96;              // D = H = W
constexpr int kC      = 3;
constexpr int kTD     = 4;               // tile extent along D
constexpr int kTH     = 4;               // tile extent along H
constexpr int kLD     = kTD + 2;         // lines incl. halo
constexpr int kLH     = kTH + 2;
constexpr int kLine   = 296;             // [pad1][halo -1: 3][payload 288][halo 96: 3][pad1]
constexpr int kLines  = kLD * kLH;       // 36
constexpr int kLdsFloats = kLines * kLine;    // 10656 floats -> ~42 KB
constexpr int kThreads = 256;            // 8 waves (wave32)
constexpr int kLineBytes = kN * kC * 4;  // 1152 B global payload per line (16B multiple)
constexpr int kChunks  = kLineBytes / 16;     // 72 x b128 per line
}

__global__ __launch_bounds__(kThreads)
void gauss3d_async_kernel(const float* __restrict__ field,
                          const float* __restrict__ log_sigma,
                          float* __restrict__ out)
{
    __shared__ __align__(16) float tile[kLdsFloats];
    const int tid = threadIdx.x;
    const int ht  = blockIdx.x;
    const int dt  = blockIdx.y;
    const int b   = blockIdx.z;
    const int d0  = dt * kTD;
    const int h0  = ht * kTH;

    const bool boundary = (dt == 0) || (dt == kN / kTD - 1) ||
                          (ht == 0) || (ht == kN / kTH - 1);

    // ---- prefetch the tile's log_sigma lines while we set up LDS ----
    {
        const int lsbase = ((b * kN + d0) * kN + h0) * kN;
        for (int i = tid; i < kTD * kTH * (kN / 32); i += kThreads) {
            int blk = i % (kN / 32);          // 3 x 128B per line
            int r   = i / (kN / 32);
            int hh  = r % kTH;
            int dd  = r / kTH;
            __builtin_prefetch(&log_sigma[lsbase + (dd * kN + hh) * kN + blk * 32], 0, 3);
        }
    }

    // ---- zero halo (full tile only when the tile touches the volume edge) ----
    float4* t4 = (float4*)tile;
    const float4 z4 = make_float4(0.f, 0.f, 0.f, 0.f);
    if (boundary) {
        for (int i = tid; i < kLdsFloats / 4; i += kThreads) t4[i] = z4;
    } else {
        // per line: floats {0..3} (pad + w=-1 halo) and {292..295} (w=96 halo + pad)
        for (int i = tid; i < kLines * 2; i += kThreads) {
            int line = i >> 1;
            int half = i & 1;
            t4[(line * kLine + half * 292) >> 2] = z4;
        }
    }
    __syncthreads();  // DS stores drained before async engine writes LDS

    // ---- gfx1250 async global -> LDS staging: 36 lines x 72 b128 chunks ----
    const uint32_t lds_base = (uint32_t)(size_t)(&tile[0]);
    for (int e = tid; e < kLines * kChunks; e += kThreads) {
        int line  = e / kChunks;
        int chunk = e - line * kChunks;
        int ld    = line / kLH;
        int lh    = line - ld * kLH;
        int d     = d0 - 1 + ld;
        int h     = h0 - 1 + lh;
        if ((unsigned)d < (unsigned)kN && (unsigned)h < (unsigned)kN) {
            uint32_t goff = (uint32_t)(((b * kN + d) * kN + h) * kN) * (kC * 4u)
                            + (uint32_t)chunk * 16u;
            uint32_t loff = lds_base + (uint32_t)(line * kLine * 4 + 16 + chunk * 16);
            // GVS mode: lds_addr VGPR, 32-bit voffset VGPR, 64-bit SGPR base
            asm volatile("global_load_async_to_lds_b128 %0, %1, %2"
                         :
                         : "v"(loff), "v"(goff), "s"(field)
                         : "memory");
        }
    }
    asm volatile("s_wait_asynccnt 0x0" ::: "memory");  // drain ASYNCcnt
    __syncthreads();

    // ---- compute: w_k = t^{d2_k}, norm = (1+2t)^3, t = exp(-1/(2*sigma^2)) ----
    constexpr int kNV = kTD * kTH * kN;  // 1536 voxels per tile
    for (int v = tid; v < kNV; v += kThreads) {
        int w  = v % kN;
        int r  = v / kN;
        int hh = r % kTH;
        int dd = r / kTH;
        int d  = d0 + dd;
        int h  = h0 + hh;

        float ls = log_sigma[((b * kN + d) * kN + h) * kN + w];
        // sigma = exp(ls); 1/(2 sigma^2) = 0.5*exp(-2 ls); t = exp(-that)
        float t1 = expf(-0.5f * expf(-2.0f * ls));

        float s[4][3];
        #pragma unroll
        for (int q = 0; q < 4; ++q) { s[q][0] = 0.f; s[q][1] = 0.f; s[q][2] = 0.f; }

        #pragma unroll
        for (int di = 0; di < 3; ++di) {
            #pragma unroll
            for (int dj = 0; dj < 3; ++dj) {
                // voxel v stored at float 1+(v+1)*3 -> taps w-1..w+1 are lp[0..8]
                const float* lp =
                    &tile[((dd + di) * kLH + (hh + dj)) * kLine + 1 + w * kC];
                #pragma unroll
                for (int dk = 0; dk < 3; ++dk) {
                    const int d2 = (di - 1) * (di - 1) + (dj - 1) * (dj - 1) +
                                   (dk - 1) * (dk - 1);
                    s[d2][0] += lp[dk * 3 + 0];
                    s[d2][1] += lp[dk * 3 + 1];
                    s[d2][2] += lp[dk * 3 + 2];
                }
            }
        }

        float t2  = t1 * t1;
        float t3  = t2 * t1;
        float nrm = 1.0f + 2.0f * t1;
        float inv = 1.0f / (nrm * nrm * nrm);
        size_t o = ((((size_t)b * kN + d) * kN + h) * kN + w) * (size_t)kC;
        out[o + 0] = (s[0][0] + t1 * s[1][0] + t2 * s[2][0] + t3 * s[3][0]) * inv;
        out[o + 1] = (s[0][1] + t1 * s[1][1] + t2 * s[2][1] + t3 * s[3][1]) * inv;
        out[o + 2] = (s[0][2] + t1 * s[1][2] + t2 * s[2][2] + t3 * s[3][2]) * inv;
    }
}

extern "C" void kernel_launch(void* const* d_in, const int* in_sizes, int n_in,
                              void* d_out, int out_size, void* d_ws, size_t ws_size,
                              hipStream_t stream) {
    (void)in_sizes; (void)n_in; (void)out_size; (void)d_ws; (void)ws_size;
    const float* field     = (const float*)d_in[0];
    const float* log_sigma = (const float*)d_in[1];
    float*       out       = (float*)d_out;
    dim3 grid(kN / kTH, kN / kTD, kB);   // 24 x 24 x 2 = 1152 workgroups
    gauss3d_async_kernel<<<grid, dim3(kThreads), 0, stream>>>(field, log_sigma, out);
}